// RPNPostProcessor_75273596829696
// MI455X (gfx1250) — compile-verified
//
#include <hip/hip_runtime.h>
#include <hip/hip_bf16.h>
#include <stdint.h>

// ---------------- problem constants (from reference setup_inputs) ----------
#define NIMG   8
#define AA     3
#define HH     256
#define WW     320
#define KTOT   (AA * HH * WW)     // 245760 anchors per image (240 * 1024)
#define PRE_N  2000
#define POST_N 1000
#define BLK    1024               // 32 wave32 waves per workgroup (1 WGP)
#define BOX_STRIDE 8192           // padded per-image box row (floats)
#define SEL_STRIDE 2048

#define NMS_THRESH      0.7f
#define TO_REMOVE       1.0f
#define BBOX_XFORM_CLIP 4.135166556742356f   // log(1000/16)

typedef unsigned int u32;
typedef __attribute__((ext_vector_type(4))) unsigned int u32x4;
typedef __attribute__((ext_vector_type(8))) int          i32x8;
typedef __attribute__((ext_vector_type(4))) int          i32x4;

// Order-preserving float -> u32 key (monotone increasing).
__device__ __forceinline__ u32 orderKey(float x) {
  u32 b = __float_as_uint(x);
  return (b & 0x80000000u) ? ~b : (b | 0x80000000u);
}
__device__ __forceinline__ float keyToFloat(u32 key) {
  u32 b = (key & 0x80000000u) ? (key & 0x7FFFFFFFu) : ~key;
  return __uint_as_float(b);
}

// score-index k (NHWA flatten) -> objectness logit (NAHW layout) -> key
__device__ __forceinline__ u32 keyAt(const float* __restrict__ obj, int n, u32 k) {
  u32 a = k % AA;
  u32 r = k / AA;
  u32 w = r % WW;
  u32 h = r / WW;
  float v = obj[((((u32)n * AA + a) * HH + h) * WW) + w];
  return orderKey(v);
}

// ======================= K1: exact top-2000 selection ======================
// Base-256 radix select (4 LDS-histogram passes) finds the exact 2000th-
// largest key incl. JAX lowest-index tie-breaking, then one stable
// compaction pass (wave32 ballot scan, 3 barriers/chunk) emits the 2000
// (key, idx) pairs in index order.
__global__ void __launch_bounds__(BLK)
k1_topk_select(const float* __restrict__ obj,
               u32* __restrict__ selKey, u32* __restrict__ selIdx) {
  const int n    = blockIdx.x;
  const int tid  = threadIdx.x;
  const u32 lane = (u32)tid & 31u;
  const u32 wid  = (u32)tid >> 5;
  __shared__ u32 hist[256];
  __shared__ u32 waveTot[32];
  __shared__ u32 sBucket, sWant;

  // ---- 4-level radix select: T = 2000th largest key ----------------------
  u32 prefHi = 0;          // resolved high bits of T (above current byte)
  u32 want   = PRE_N;      // rank of T among keys matching prefHi (1-based)
  for (int L = 0; L < 4; ++L) {
    const int shift = 24 - 8 * L;
    for (int v = tid; v < 256; v += BLK) hist[v] = 0;
    __syncthreads();
    for (u32 k = tid; k < KTOT; k += BLK) {
      u32 key = keyAt(obj, n, k);
      bool match = (L == 0) || ((key >> (shift + 8)) == prefHi);
      if (match) atomicAdd(&hist[(key >> shift) & 0xFFu], 1u);
    }
    __syncthreads();
    if (tid == 0) {
      u32 cum = 0, b = 0;
      for (int v = 255; v >= 0; --v) {          // descending cumulative scan
        u32 h = hist[v];
        if (cum + h >= want) { b = (u32)v; break; }
        cum += h;
      }
      sBucket = b;
      sWant   = want - cum;                     // rank within chosen bucket
    }
    __syncthreads();
    prefHi = (prefHi << 8) | sBucket;
    want   = sWant;
    __syncthreads();
  }
  const u32 T   = prefHi;          // exact 2000th-largest key
  const u32 e   = want;            // number of ==T ties to take (by index)
  const u32 cGT = PRE_N - e;       // count of keys strictly greater than T

  // ---- stable compaction: wave ballot rank + 32-entry inter-wave prefix --
  __shared__ u32 baseGT, baseEQ;
  if (tid == 0) { baseGT = 0; baseEQ = 0; }
  __syncthreads();

  u32* sk = selKey + (size_t)n * SEL_STRIDE;
  u32* si = selIdx + (size_t)n * SEL_STRIDE;
  const u32 below = (1u << lane) - 1u;

  for (u32 c0 = 0; c0 < KTOT; c0 += BLK) {
    u32 k   = c0 + tid;
    u32 key = (k < KTOT) ? keyAt(obj, n, k) : 0u;
    u32 gt  = (k < KTOT && key >  T) ? 1u : 0u;
    u32 eq  = (k < KTOT && key == T) ? 1u : 0u;

    u32 mgt = (u32)__ballot(gt != 0u);
    u32 meq = (u32)__ballot(eq != 0u);
    u32 rgt = __popc(mgt & below);              // stable intra-wave rank
    u32 req = __popc(meq & below);
    if (lane == 0)
      waveTot[wid] = ((u32)__popc(mgt) << 16) | (u32)__popc(meq);
    __syncthreads();

    u32 preGT = 0, preEQ = 0, totGT = 0, totEQ = 0;
#pragma unroll
    for (int w = 0; w < 32; ++w) {              // LDS broadcast reads
      u32 v = waveTot[w];
      totGT += v >> 16;   totEQ += v & 0xFFFFu;
      if (w < (int)wid) { preGT += v >> 16; preEQ += v & 0xFFFFu; }
    }
    u32 bGT = baseGT, bEQ = baseEQ;

    if (gt) {
      u32 slot = bGT + preGT + rgt;
      sk[slot] = key; si[slot] = k;
    }
    if (eq) {
      u32 rEQ = bEQ + preEQ + req;
      if (rEQ < e) {                            // first e ties by index
        u32 slot = cGT + rEQ;
        sk[slot] = key; si[slot] = k;
      }
    }
    __syncthreads();
    if (tid == 0) { baseGT += totGT; baseEQ += totEQ; }
    __syncthreads();
  }
}

// ================= K2: LDS bitonic sort + decode + clip ====================
__global__ void __launch_bounds__(BLK)
k2_sort_decode(const float* __restrict__ anchors,
               const float* __restrict__ obj,
               const float* __restrict__ breg,
               const int*   __restrict__ ishapes,
               const u32* __restrict__ selKey, const u32* __restrict__ selIdx,
               float* __restrict__ boxes, float* __restrict__ scores) {
  const int n   = blockIdx.x;
  const int tid = threadIdx.x;
  __shared__ u32 sk[2048];
  __shared__ u32 si[2048];

  for (int r = tid; r < 2048; r += BLK) {
    if (r < PRE_N) {
      sk[r] = selKey[(size_t)n * SEL_STRIDE + r];
      si[r] = selIdx[(size_t)n * SEL_STRIDE + r];
    } else {
      sk[r] = 0u;             // padded sentinels sink to the back
      si[r] = 0xFFFFFFFFu;
    }
  }
  __syncthreads();

  // bitonic sort, descending by key, ties ascending by index (JAX stable)
  for (unsigned size = 2; size <= 2048; size <<= 1) {
    for (unsigned stride = size >> 1; stride > 0; stride >>= 1) {
      for (unsigned t = tid; t < 2048; t += BLK) {
        unsigned j = t ^ stride;
        if (j > t) {
          bool desc = ((t & size) == 0);
          u32 ka = sk[t], kb = sk[j], ia = si[t], ib = si[j];
          bool jFirst = (kb > ka) || (kb == ka && ib < ia);
          if (desc ? jFirst : !jFirst) {
            sk[t] = kb; sk[j] = ka; si[t] = ib; si[j] = ia;
          }
        }
      }
      __syncthreads();
    }
  }

  const float wmaxv = (float)ishapes[n * 2 + 0] - TO_REMOVE;
  const float hmaxv = (float)ishapes[n * 2 + 1] - TO_REMOVE;
  const size_t plane = (size_t)HH * WW;
  const float* bp = breg + (size_t)n * AA * 4 * plane;

  for (int r = tid; r < PRE_N; r += BLK) {
    u32 k = si[r];
    const float* anc = anchors + ((size_t)n * KTOT + k) * 4;
    float a0 = anc[0], a1 = anc[1], a2 = anc[2], a3 = anc[3];

    u32 a  = k % AA;
    u32 rs = k / AA;
    u32 wq = rs % WW;
    u32 hq = rs / WW;
    size_t pix = (size_t)hq * WW + wq;
    float dx = bp[((size_t)(a * 4 + 0)) * plane + pix];
    float dy = bp[((size_t)(a * 4 + 1)) * plane + pix];
    float dw = bp[((size_t)(a * 4 + 2)) * plane + pix];
    float dh = bp[((size_t)(a * 4 + 3)) * plane + pix];
    dw = fminf(dw, BBOX_XFORM_CLIP);
    dh = fminf(dh, BBOX_XFORM_CLIP);

    float w  = a2 - a0 + TO_REMOVE;
    float h  = a3 - a1 + TO_REMOVE;
    float cx = a0 + 0.5f * w;
    float cy = a1 + 0.5f * h;
    float pcx = dx * w + cx;
    float pcy = dy * h + cy;
    float pw  = __expf(dw) * w;   // fast exp; tolerance-level accurate
    float ph  = __expf(dh) * h;

    float x1 = fminf(fmaxf(pcx - 0.5f * pw, 0.f), wmaxv);
    float y1 = fminf(fmaxf(pcy - 0.5f * ph, 0.f), hmaxv);
    float x2 = fminf(fmaxf(pcx + 0.5f * pw - 1.f, 0.f), wmaxv);
    float y2 = fminf(fmaxf(pcy + 0.5f * ph - 1.f, 0.f), hmaxv);

    float* bo = boxes + (size_t)n * BOX_STRIDE + (size_t)r * 4;
    bo[0] = x1; bo[1] = y1; bo[2] = x2; bo[3] = y2;

    float logit = keyToFloat(sk[r]);
    scores[(size_t)n * SEL_STRIDE + r] = 1.0f / (1.0f + __expf(-logit));
  }
}

// ====== K3: TDM-stage boxes into LDS, greedy NMS, compact + emit ==========
__global__ void __launch_bounds__(BLK)
k3_nms_emit(const float* __restrict__ boxesG,
            const float* __restrict__ scoresG,
            float* __restrict__ out) {
  const int n    = blockIdx.x;
  const int tid  = threadIdx.x;
  const u32 lane = (u32)tid & 31u;
  const u32 wid  = (u32)tid >> 5;
  __shared__ __align__(16) float sb[BOX_STRIDE]; // 32 KB box tile in LDS
  __shared__ unsigned char  sKeep[2048];
  __shared__ unsigned char  sSup[2048];
  __shared__ unsigned short sPos[2048];
  __shared__ u32            wtot[64];
  __shared__ int            sCur;
  __shared__ int            sTotKeep;

  // --- CDNA5 Tensor Data Mover: 1 DMA descriptor loads 2000x4 f32 to LDS --
  if ((tid >> 5) == 0) {   // wave 0 only issues the TDM op
    unsigned long long ga = (unsigned long long)(uintptr_t)(boxesG + (size_t)n * BOX_STRIDE);
    u32 ldsAddr = (u32)(uintptr_t)&sb[0];        // LDS byte offset (addr[31:0])
    u32x4 g0;
    g0.x = 1u;                                             // count=1, user mode
    g0.y = ldsAddr;                                        // lds_addr
    g0.z = (u32)(ga & 0xFFFFFFFFull);                      // global_addr[31:0]
    g0.w = (u32)((ga >> 32) & 0x01FFFFFFull) | (2u << 30); // addr[56:32] | type=2
    i32x8 g1 = {
      (int)(2u << 16),            // data_size = 4B
      (int)(8000u << 16),         // tensor_dim0[15:0] = 8000
      (int)(1u  << 16),           // tensor_dim0 hi = 0 | tensor_dim1 = 1
      (int)(8000u << 16),         // tensor_dim1 hi = 0 | tile_dim0 = 8000
      1,                          // tile_dim1 = 1, tile_dim2 = 0
      8000,                       // tensor_dim0_stride lo
      0, 0
    };
    i32x4 gz = {0, 0, 0, 0};
#if __clang_major__ >= 23
    i32x8 gz8 = {0, 0, 0, 0, 0, 0, 0, 0};
    __builtin_amdgcn_tensor_load_to_lds(g0, g1, gz, gz, gz8, 0);
#else
    __builtin_amdgcn_tensor_load_to_lds(g0, g1, gz, gz, 0);
#endif
    __builtin_amdgcn_s_wait_tensorcnt(0);
  }
  __syncthreads();   // DMA complete before any LDS state is initialized

  // belt-and-braces: cooperative b128 load of the same tile (identical
  // values; benign overwrite) -- D# layout is ISA-derived, not HW-verified
  {
    const float4* s4 = reinterpret_cast<const float4*>(boxesG + (size_t)n * BOX_STRIDE);
    float4*       d4 = reinterpret_cast<float4*>(sb);
    for (int r = tid; r < PRE_N; r += BLK) d4[r] = s4[r];
  }
  for (int r = tid; r < 2048; r += BLK) { sSup[r] = 0; sKeep[r] = 0; }
  // warm caches for the final score gather while the NMS loop runs
  for (int r = tid; r < PRE_N; r += BLK)
    __builtin_prefetch(scoresG + (size_t)n * SEL_STRIDE + r, 0, 1);
  __syncthreads();

  // --- greedy NMS: serial keep decision, parallel suppression -------------
  for (int i = 0; i < PRE_N; ++i) {
    if (tid == 0) {
      float x1 = sb[i*4], y1 = sb[i*4+1], x2 = sb[i*4+2], y2 = sb[i*4+3];
      bool valid = ((x2 - x1 + TO_REMOVE) >= 0.0f) && ((y2 - y1 + TO_REMOVE) >= 0.0f);
      int kp = (valid && !sSup[i]) ? 1 : 0;
      sKeep[i] = (unsigned char)kp;
      sCur = kp;
    }
    __syncthreads();
    if (sCur) {
      float x1i = sb[i*4], y1i = sb[i*4+1], x2i = sb[i*4+2], y2i = sb[i*4+3];
      float ai = fmaxf(x2i - x1i + TO_REMOVE, 0.f) * fmaxf(y2i - y1i + TO_REMOVE, 0.f);
      for (int j = i + 1 + tid; j < PRE_N; j += BLK) {
        float x1j = sb[j*4], y1j = sb[j*4+1], x2j = sb[j*4+2], y2j = sb[j*4+3];
        float aj = fmaxf(x2j - x1j + TO_REMOVE, 0.f) * fmaxf(y2j - y1j + TO_REMOVE, 0.f);
        float xx1 = fmaxf(x1i, x1j), yy1 = fmaxf(y1i, y1j);
        float xx2 = fminf(x2i, x2j), yy2 = fminf(y2i, y2j);
        float inter = fmaxf(xx2 - xx1 + TO_REMOVE, 0.f) * fmaxf(yy2 - yy1 + TO_REMOVE, 0.f);
        float iou = inter / (ai + aj - inter + 1e-9f);
        if (iou > NMS_THRESH) sSup[j] = 1;
      }
    }
    __syncthreads();
  }

  // --- stable block prefix of keep flags via wave32 ballots ---------------
  {
    u32 p0 = (tid         < PRE_N) ? (u32)sKeep[tid]        : 0u;
    u32 p1 = (tid + 1024  < PRE_N) ? (u32)sKeep[tid + 1024] : 0u;
    u32 m0 = (u32)__ballot(p0 != 0u);
    u32 m1 = (u32)__ballot(p1 != 0u);
    if (lane == 0) { wtot[wid] = (u32)__popc(m0); wtot[wid + 32] = (u32)__popc(m1); }
    __syncthreads();
    u32 below = (1u << lane) - 1u;
    u32 pre0 = 0, pre1 = 0, tot = 0;
#pragma unroll
    for (int w = 0; w < 64; ++w) {
      u32 v = wtot[w];
      if (w < (int)wid)        pre0 += v;
      if (w < (int)(wid + 32)) pre1 += v;
      tot += v;
    }
    sPos[tid]        = (unsigned short)(pre0 + __popc(m0 & below));
    sPos[tid + 1024] = (unsigned short)(pre1 + __popc(m1 & below));
    if (tid == 0) sTotKeep = (int)tot;
  }
  __syncthreads();

  float* obx = out;                                        // (N, 1000, 4)
  float* osc = out + (size_t)NIMG * POST_N * 4;            // (N, 1000)
  float* ovl = osc + (size_t)NIMG * POST_N;                // (N, 1000) bool->float
  float* bx  = obx + (size_t)n * POST_N * 4;
  float* sc  = osc + (size_t)n * POST_N;
  float* vl  = ovl + (size_t)n * POST_N;

  // zero only slots beyond keepCount (disjoint from kept writes below)
  for (int s = tid; s < POST_N; s += BLK) {
    if (s >= sTotKeep) {
      bx[s*4+0] = 0.f; bx[s*4+1] = 0.f; bx[s*4+2] = 0.f; bx[s*4+3] = 0.f;
      sc[s] = 0.f; vl[s] = 0.f;
    }
  }
  for (int r = tid; r < PRE_N; r += BLK) {
    if (sKeep[r]) {
      int s = (int)sPos[r];
      if (s < POST_N) {
        bx[s*4+0] = sb[r*4+0]; bx[s*4+1] = sb[r*4+1];
        bx[s*4+2] = sb[r*4+2]; bx[s*4+3] = sb[r*4+3];
        sc[s] = scoresG[(size_t)n * SEL_STRIDE + r];
        vl[s] = 1.0f;
      }
    }
  }
}

// ============================ host-side launch =============================
extern "C" void kernel_launch(void* const* d_in, const int* in_sizes, int n_in,
                              void* d_out, int out_size, void* d_ws, size_t ws_size,
                              hipStream_t stream) {
  const float* anchors = (const float*)d_in[0];  // (N, K, 4)
  const float* obj     = (const float*)d_in[1];  // (N, A, H, W)
  const float* breg    = (const float*)d_in[2];  // (N, A*4, H, W)
  const int*   ishapes = (const int*)  d_in[3];  // (N, 2)

  // workspace layout (u32/f32 units), ~448 KB total
  u32*   selKey = (u32*)d_ws;
  u32*   selIdx = selKey + (size_t)NIMG * SEL_STRIDE;
  float* boxes  = (float*)(selIdx + (size_t)NIMG * SEL_STRIDE);
  float* scores = boxes + (size_t)NIMG * BOX_STRIDE;

  dim3 grid(NIMG), block(BLK);
  k1_topk_select<<<grid, block, 0, stream>>>(obj, selKey, selIdx);
  k2_sort_decode<<<grid, block, 0, stream>>>(anchors, obj, breg, ishapes,
                                             selKey, selIdx, boxes, scores);
  k3_nms_emit<<<grid, block, 0, stream>>>(boxes, scores, (float*)d_out);
}